// RKDLoss_45672682225799
// MI455X (gfx1250) — compile-verified
//
#include <hip/hip_runtime.h>
#include <math.h>

#define BDIM 512
#define DDIM 256
#define BB   (BDIM * BDIM)
#define ITILE 8

typedef __attribute__((ext_vector_type(2))) float v2f;
typedef __attribute__((ext_vector_type(8))) float v8f;

// ---------------------------------------------------------------- init
__global__ void rkd_init_accum(double* accum) {
    if (threadIdx.x < 4) accum[threadIdx.x] = 0.0;
}

// ---------------------------------------------------------------- gram: P = E @ E^T via f32 WMMA
// grid = (32, 32, 2): x = N tile, y = M tile, z = {student, teacher}; block = 32 (one wave)
__global__ __launch_bounds__(32) void rkd_gram_wmma(const float* __restrict__ S,
                                                    const float* __restrict__ T,
                                                    float* __restrict__ P) {
    const float* E  = (blockIdx.z == 0) ? S : T;
    float*       Pm = P + (size_t)blockIdx.z * BB;

    const int lane = threadIdx.x & 31;
    const int mrow = (blockIdx.y << 4) + (lane & 15);
    const int nrow = (blockIdx.x << 4) + (lane & 15);
    const int koff = (lane >> 4) << 1;   // lanes 0-15 -> K+0/K+1, lanes 16-31 -> K+2/K+3

    const float* arow = E + (size_t)mrow * DDIM + koff;
    const float* brow = E + (size_t)nrow * DDIM + koff;

    v8f acc = {};
    #pragma unroll 4
    for (int k = 0; k < DDIM; k += 4) {
        v2f a, b;
        a.x = arow[k];     a.y = arow[k + 1];
        b.x = brow[k];     b.y = brow[k + 1];
        acc = __builtin_amdgcn_wmma_f32_16x16x4_f32(false, a, false, b,
                                                    (short)0, acc, false, false);
    }

    // C/D layout: VGPR r holds M = (lane/16)*8 + r, N = lane%16
    const int cn = (blockIdx.x << 4) + (lane & 15);
    const int cm = (blockIdx.y << 4) + ((lane >> 4) << 3);
    #pragma unroll
    for (int r = 0; r < 8; ++r)
        Pm[(size_t)(cm + r) * BDIM + cn] = acc[r];
}

// ---------------------------------------------------------------- distances, invn tables, sum(d)
__global__ __launch_bounds__(256) void rkd_dist_invn(const float* __restrict__ P,
                                                     float* __restrict__ invn,
                                                     double* __restrict__ accum) {
    const int idx = blockIdx.x * 256 + threadIdx.x;   // idx = i*512 + j
    const int i = idx >> 9, j = idx & 511;

    float sums[2];
    #pragma unroll
    for (int z = 0; z < 2; ++z) {
        const float* Pm = P + (size_t)z * BB;
        const float pii = Pm[(size_t)i * BDIM + i];
        const float pjj = Pm[(size_t)j * BDIM + j];
        const float pij = Pm[idx];
        const float sq  = pii + pjj - 2.0f * pij;
        sums[z] = sqrtf(fmaxf(sq, 1e-12f));           // clip(1e-12) -> all entries > 0
        const float nrm = sqrtf(fmaxf(sq, 0.0f));
        invn[(size_t)z * BB + idx] = (i == j) ? 0.0f : 1.0f / fmaxf(nrm, 1e-12f);
    }

    __shared__ float red[256];
    #pragma unroll
    for (int z = 0; z < 2; ++z) {
        red[threadIdx.x] = sums[z];
        __syncthreads();
        for (int s = 128; s > 0; s >>= 1) {
            if (threadIdx.x < s) red[threadIdx.x] += red[threadIdx.x + s];
            __syncthreads();
        }
        if (threadIdx.x == 0) atomicAdd(&accum[z], (double)red[0]);
        __syncthreads();
    }
}

// ---------------------------------------------------------------- distance smooth-L1
__global__ __launch_bounds__(256) void rkd_dist_loss(const float* __restrict__ P,
                                                     double* __restrict__ accum) {
    const int idx = blockIdx.x * 256 + threadIdx.x;
    const int i = idx >> 9, j = idx & 511;

    float dv[2];
    #pragma unroll
    for (int z = 0; z < 2; ++z) {
        const float* Pm = P + (size_t)z * BB;
        const float sq = Pm[(size_t)i * BDIM + i] + Pm[(size_t)j * BDIM + j]
                       - 2.0f * Pm[idx];
        dv[z] = sqrtf(fmaxf(sq, 1e-12f));
    }
    const float mean_s = (float)(accum[0] * (1.0 / (double)BB));
    const float mean_t = (float)(accum[1] * (1.0 / (double)BB));
    const float s_d = dv[0] / (mean_s + 1e-8f);
    const float t_d = dv[1] / (mean_t + 1e-8f);
    const float diff = fabsf(s_d - t_d);
    const float l = (diff < 1.0f) ? 0.5f * diff * diff : diff - 0.5f;

    __shared__ float red[256];
    red[threadIdx.x] = l;
    __syncthreads();
    for (int s = 128; s > 0; s >>= 1) {
        if (threadIdx.x < s) red[threadIdx.x] += red[threadIdx.x + s];
        __syncthreads();
    }
    if (threadIdx.x == 0) atomicAdd(&accum[2], (double)red[0]);
}

// ---------------------------------------------------------------- angle smooth-L1
// Rank-structured B^3 pass with 8-way i-tiling + (j,k) symmetry (k >= j, weight 2 off-diag).
// grid = (8, 64): x = 64-row j chunk, y = 8-row i tile; block = 512 (k = threadIdx.x)
__global__ __launch_bounds__(512) void rkd_angle_loss(const float* __restrict__ P,
                                                      const float* __restrict__ invn,
                                                      double* __restrict__ accum) {
    // per column c of i-row ii: {P_s[i][c], P_t[i][c], invn_s[i][c], invn_t[i][c]}
    __shared__ float4 sJ[ITILE][BDIM];           // 64 KB
    __shared__ float  red[512];                  // 2 KB

    const int k  = threadIdx.x;                  // 0..511
    const int j0 = blockIdx.x * 64;
    const int i0 = blockIdx.y * ITILE;

    const float* Ps = P;
    const float* Pt = P + BB;
    const float* Is = invn;
    const float* It = invn + BB;

    for (int t = threadIdx.x; t < ITILE * BDIM; t += 512) {
        const int ii = t >> 9, c = t & 511;
        const size_t o = (size_t)(i0 + ii) * BDIM + c;
        float4 v;
        v.x = Ps[o];  v.y = Pt[o];  v.z = Is[o];  v.w = It[o];
        sJ[ii][c] = v;
    }
    __syncthreads();

    // per-thread register preload: i-row values at this thread's own k, plus P[i,i]
    float rPs[ITILE], rPt[ITILE], rIs[ITILE], rIt[ITILE], pS[ITILE], pT[ITILE];
    #pragma unroll
    for (int ii = 0; ii < ITILE; ++ii) {
        const float4 rk = sJ[ii][k];
        rPs[ii] = rk.x;  rPt[ii] = rk.y;  rIs[ii] = rk.z;  rIt[ii] = rk.w;
        const float4 dg = sJ[ii][i0 + ii];
        pS[ii] = dg.x;   pT[ii] = dg.y;
    }

    float acc = 0.0f;
    for (int jj = 0; jj < 64; ++jj) {
        const int j = j0 + jj;
        if (k >= j) {
            const float pjs = Ps[(size_t)j * BDIM + k];
            const float pjt = Pt[(size_t)j * BDIM + k];
            float lsum = 0.0f;
            #pragma unroll
            for (int ii = 0; ii < ITILE; ++ii) {
                const float4 vj = sJ[ii][j];     // broadcast ds_load_b128
                const float sa = (pjs + (pS[ii] - vj.x) - rPs[ii]) * vj.z * rIs[ii];
                const float ta = (pjt + (pT[ii] - vj.y) - rPt[ii]) * vj.w * rIt[ii];
                const float diff = fabsf(sa - ta);
                lsum += (diff < 1.0f) ? 0.5f * diff * diff : diff - 0.5f;
            }
            acc += (k == j) ? lsum : 2.0f * lsum;
        }
    }

    red[threadIdx.x] = acc;
    __syncthreads();
    for (int s = 256; s > 0; s >>= 1) {
        if (threadIdx.x < s) red[threadIdx.x] += red[threadIdx.x + s];
        __syncthreads();
    }
    if (threadIdx.x == 0) atomicAdd(&accum[3], (double)red[0]);
}

// ---------------------------------------------------------------- finalize
__global__ void rkd_finalize(const double* __restrict__ accum, float* __restrict__ out) {
    if (threadIdx.x == 0 && blockIdx.x == 0) {
        const double loss_d = accum[2] / (double)BB;
        const double loss_a = accum[3] / ((double)BDIM * (double)BDIM * (double)BDIM);
        out[0] = (float)(1.0 * loss_d + 2.0 * loss_a);
    }
}

extern "C" void kernel_launch(void* const* d_in, const int* in_sizes, int n_in,
                              void* d_out, int out_size, void* d_ws, size_t ws_size,
                              hipStream_t stream) {
    (void)in_sizes; (void)n_in; (void)out_size; (void)ws_size;
    const float* S = (const float*)d_in[0];   // student [512, 256] f32
    const float* T = (const float*)d_in[1];   // teacher [512, 256] f32

    float*  P     = (float*)d_ws;             // 2 * 512*512 f32 grams (student, teacher)
    float*  invn  = P + 2 * BB;               // 2 * 512*512 f32 inverse-norm tables
    double* accum = (double*)(invn + 2 * BB); // [sum_d_s, sum_d_t, loss_d_sum, loss_a_sum]

    rkd_init_accum<<<1, 32, 0, stream>>>(accum);
    rkd_gram_wmma<<<dim3(32, 32, 2), 32, 0, stream>>>(S, T, P);
    rkd_dist_invn<<<BB / 256, 256, 0, stream>>>(P, invn, accum);
    rkd_dist_loss<<<BB / 256, 256, 0, stream>>>(P, accum);
    rkd_angle_loss<<<dim3(8, 64), 512, 0, stream>>>(P, invn, accum);
    rkd_finalize<<<1, 32, 0, stream>>>(accum, (float*)d_out);
}